// AttentionNodeTypeSheafLearner_31842887533263
// MI455X (gfx1250) — compile-verified
//
#include <hip/hip_runtime.h>
#include <math.h>

// ---------------------------------------------------------------------------
// AttentionNodeTypeSheafLearner, MI455X (gfx1250, wave32)
//
// Insight: edge features depend only on (type[src], type[dst]) -> 64 distinct
// rows. Kernel 1 computes the full 64x16 output table with V_WMMA_F32_16X16X4
// (exact f32 matmul). Kernel 2 is the bandwidth-bound path: per-edge type
// lookup + 64B table-row write (~230 MB total => ~10us at 23.3 TB/s).
// ---------------------------------------------------------------------------

typedef __attribute__((ext_vector_type(2))) float v2f;
typedef __attribute__((ext_vector_type(8))) float v8f;

#define TIN 16      // 2 * NUM_NODE_TYPES
#define HID 64
#define NT 8
#define NPAIR 64    // NT*NT table rows

// ---------------------------------------------------------------------------
// Kernel 1: one block, 4 waves. Build table[64][16] = eye(4) - softmax(MLP(h)).
// Wave w owns M-tile rows [16w, 16w+16).
// f32 WMMA fragment layouts (ISA 7.12.2):
//   A 16x4 : lanes 0-15 -> {K0,K1} in v[0],v[1]; lanes 16-31 -> {K2,K3}
//   B 4x16 : v[0]: K0(lanes0-15)/K2(lanes16-31); v[1]: K1/K3; N = lane&15
//   C 16x16: v[r]: row r (lanes0-15) / row r+8 (lanes16-31); N = lane&15
// ---------------------------------------------------------------------------
__global__ __launch_bounds__(128) void sheaf_table_kernel(
    const float* __restrict__ ln_g, const float* __restrict__ ln_b,
    const float* __restrict__ W1, const float* __restrict__ b1,
    const float* __restrict__ W2, const float* __restrict__ b2,
    float* __restrict__ table)
{
    __shared__ float sH [NPAIR][TIN];   // layernormed one-hot rows
    __shared__ float sW1[TIN][HID];
    __shared__ float sB1[HID];
    __shared__ float sW2[HID][TIN];
    __shared__ float sB2[TIN];
    __shared__ float sH1[NPAIR][HID];   // relu(H @ W1 + b1)
    __shared__ float sO [NPAIR][TIN];   // H1 @ W2 + b2 (pre-softmax)

    const int tid = threadIdx.x;

    for (int i = tid; i < TIN * HID; i += 128) sW1[i >> 6][i & 63] = W1[i];
    for (int i = tid; i < HID * TIN; i += 128) sW2[i >> 4][i & 15] = W2[i];
    if (tid < HID) sB1[tid] = b1[tid];
    if (tid < TIN) sB2[tid] = b2[tid];

    if (tid < NPAIR) {
        const int ts = tid >> 3, tt = tid & 7;
        // h has exactly two ones -> LN stats are compile-time constants
        const float mu = 2.0f / 16.0f;
        const float rs = rsqrtf(0.109375f + 1e-5f);
        for (int j = 0; j < TIN; ++j) {
            float h = (j < 8) ? ((j == ts) ? 1.0f : 0.0f)
                              : ((j - 8 == tt) ? 1.0f : 0.0f);
            sH[tid][j] = (h - mu) * rs * ln_g[j] + ln_b[j];
        }
    }
    __syncthreads();

    const int wave = tid >> 5;
    const int lane = tid & 31;
    const int half = lane >> 4;   // 0: lanes 0-15, 1: lanes 16-31
    const int l15  = lane & 15;
    const int m0   = wave * 16;

    // GEMM1: sH1 = relu(sH @ sW1 + b1)   [64x16]@[16x64]
    for (int nt = 0; nt < 4; ++nt) {
        const int n0 = nt * 16;
        v8f acc = {};
        for (int kk = 0; kk < 4; ++kk) {
            const int k0 = kk * 4;
            v2f a, b;
            a.x = sH[m0 + l15][k0 + 2 * half + 0];
            a.y = sH[m0 + l15][k0 + 2 * half + 1];
            b.x = sW1[k0 + 2 * half + 0][n0 + l15];
            b.y = sW1[k0 + 2 * half + 1][n0 + l15];
            acc = __builtin_amdgcn_wmma_f32_16x16x4_f32(
                false, a, false, b, (short)0, acc, false, false);
        }
        for (int r = 0; r < 8; ++r) {
            float v = acc[r] + sB1[n0 + l15];
            sH1[m0 + r + 8 * half][n0 + l15] = v > 0.0f ? v : 0.0f;
        }
    }
    __syncthreads();

    // GEMM2: sO = sH1 @ sW2 + b2   [64x64]@[64x16]
    {
        v8f acc = {};
        for (int kk = 0; kk < 16; ++kk) {
            const int k0 = kk * 4;
            v2f a, b;
            a.x = sH1[m0 + l15][k0 + 2 * half + 0];
            a.y = sH1[m0 + l15][k0 + 2 * half + 1];
            b.x = sW2[k0 + 2 * half + 0][l15];
            b.y = sW2[k0 + 2 * half + 1][l15];
            acc = __builtin_amdgcn_wmma_f32_16x16x4_f32(
                false, a, false, b, (short)0, acc, false, false);
        }
        for (int r = 0; r < 8; ++r)
            sO[m0 + r + 8 * half][l15] = acc[r] + sB2[l15];
    }
    __syncthreads();

    // row-wise 4x4 softmax, output = eye(4) - softmax
    if (tid < NPAIR) {
        for (int g = 0; g < 4; ++g) {
            float v[4];
            for (int i = 0; i < 4; ++i) v[i] = sO[tid][4 * g + i];
            float m = fmaxf(fmaxf(v[0], v[1]), fmaxf(v[2], v[3]));
            float e[4], s = 0.0f;
            for (int i = 0; i < 4; ++i) { e[i] = __expf(v[i] - m); s += e[i]; }
            float inv = 1.0f / s;
            for (int i = 0; i < 4; ++i)
                table[tid * 16 + 4 * g + i] = ((i == g) ? 1.0f : 0.0f) - e[i] * inv;
        }
    }
}

// ---------------------------------------------------------------------------
// Kernel 2: bandwidth path. Per edge: idx = 8*type[src] + type[dst], then
// copy 64B table row (LDS-staged) to the output.  ~230 MB of traffic total.
// ---------------------------------------------------------------------------
__global__ __launch_bounds__(256) void sheaf_scatter_kernel(
    const int* __restrict__ edge_index,   // [2*E], row-major: rows then cols
    const int* __restrict__ node_types,   // [N]
    const float* __restrict__ table,      // [64*16]
    float4* __restrict__ out,             // [E*4] float4 (= E*16 floats)
    int E)
{
    __shared__ float4 sT[NPAIR * 4];      // 4 KB table
    for (int i = threadIdx.x; i < NPAIR * 4; i += blockDim.x)
        sT[i] = ((const float4*)table)[i];
    __syncthreads();

    const int stride = gridDim.x * blockDim.x;
    for (int e = blockIdx.x * blockDim.x + threadIdx.x; e < E; e += stride) {
        const int r = edge_index[e];
        const int c = edge_index[E + e];
        const int idx = (node_types[r] << 3) + node_types[c];
        const float4 t0 = sT[idx * 4 + 0];
        const float4 t1 = sT[idx * 4 + 1];
        const float4 t2 = sT[idx * 4 + 2];
        const float4 t3 = sT[idx * 4 + 3];
        out[e * 4 + 0] = t0;
        out[e * 4 + 1] = t1;
        out[e * 4 + 2] = t2;
        out[e * 4 + 3] = t3;
    }
}

// ---------------------------------------------------------------------------
// Inputs (setup_inputs order):
//  0: x [N,16] f32 (UNUSED by reference)   1: edge_index [2,E] int
//  2: node_types [N] int                    3: ln_g [16]   4: ln_b [16]
//  5: W1 [16,64]   6: b1 [64]   7: W2 [64,16]   8: b2 [16]
// Output: [E,4,4] f32.
// ---------------------------------------------------------------------------
extern "C" void kernel_launch(void* const* d_in, const int* in_sizes, int n_in,
                              void* d_out, int out_size, void* d_ws, size_t ws_size,
                              hipStream_t stream)
{
    const int*   edge_index = (const int*)d_in[1];
    const int*   node_types = (const int*)d_in[2];
    const float* ln_g = (const float*)d_in[3];
    const float* ln_b = (const float*)d_in[4];
    const float* W1   = (const float*)d_in[5];
    const float* b1   = (const float*)d_in[6];
    const float* W2   = (const float*)d_in[7];
    const float* b2   = (const float*)d_in[8];

    float* table = (float*)d_ws;            // 64*16 floats = 4 KB scratch
    const int E = in_sizes[1] / 2;

    sheaf_table_kernel<<<1, 128, 0, stream>>>(ln_g, ln_b, W1, b1, W2, b2, table);

    const int threads = 256;
    const int blocks  = (E + threads - 1) / threads;
    sheaf_scatter_kernel<<<blocks, threads, 0, stream>>>(
        edge_index, node_types, table, (float4*)d_out, E);
}